// BertEmbedding_77670188581156
// MI455X (gfx1250) — compile-verified
//
#include <hip/hip_runtime.h>
#include <hip/hip_bf16.h>

// BERT embedding gather+add, tuned for MI455X memory system:
//  - 2D grid (positions/TPB, batch) -> no scalar integer division for `pos`
//  - 8 consecutive tokens per block -> index loads coalesce into wide scalar
//    loads (KMcnt); 32 VMEM ops per thread keep the 23.3 TB/s HBM path busy
//  - single block-level bounds guard; fast path is branch-free so pos/out
//    addresses fold into immediate offsets (3072 B apart, fits 24-bit ioffset)
//  - 128-bit (b128) vector loads of the three embedding rows
//  - table loads use default RT temporal hint so the 93.8 MB word table
//    becomes resident in the 192 MB L2 (gather hits are then L2 hits)
//  - output is write-once/stream: nontemporal b128 stores (TH=NT) so the
//    50 MB output stream does not evict the embedding tables from L2
//
// H = 768 floats = 192 float4 -> blockDim = 192 (6 wave32)

#define HIDDEN 768
#define VEC_PER_ROW (HIDDEN / 4) // 192
#define TPB 8                    // tokens per block

typedef float v4f __attribute__((ext_vector_type(4)));

__device__ __forceinline__ void emit_token(
    int wid, int tid, int pos, size_t token, int h,
    const float* __restrict__ word_emb, const float* __restrict__ pos_emb,
    const float* __restrict__ type_emb, float* __restrict__ out)
{
    // Cacheable RT loads: keep embedding tables hot in L2.
    const v4f w = *reinterpret_cast<const v4f*>(
        word_emb + (size_t)wid * HIDDEN + h);
    const v4f t = *reinterpret_cast<const v4f*>(
        type_emb + (size_t)tid * HIDDEN + h);
    const v4f p = *reinterpret_cast<const v4f*>(
        pos_emb + (size_t)pos * HIDDEN + h);

    const v4f r = w + t + p;

    // Streamed, write-once output: nontemporal b128 store (TH=NT).
    __builtin_nontemporal_store(
        r, reinterpret_cast<v4f*>(out + token * HIDDEN + h));
}

__global__ __launch_bounds__(VEC_PER_ROW) void bert_embedding_kernel(
    const int* __restrict__ input_ids,        // [B*S]
    const int* __restrict__ token_type_ids,   // [B*S]
    const float* __restrict__ word_emb,       // [VOCAB, H]
    const float* __restrict__ pos_emb,        // [MAX_POS, H]
    const float* __restrict__ type_emb,       // [TYPE_VOCAB, H]
    float* __restrict__ out,                  // [B*S, H]
    int seq_len)                              // S (== 512)
{
    const int pos0    = blockIdx.x * TPB;                     // first position
    const size_t tok0 = (size_t)blockIdx.y * seq_len + pos0;  // first token
    const int h = threadIdx.x * 4;                            // float4 slot

    if (pos0 + TPB <= seq_len) {
        // ---- fast path: full tile, no per-token guards ----
        int wid[TPB], tid[TPB];
#pragma unroll
        for (int i = 0; i < TPB; ++i) {
            wid[i] = input_ids[tok0 + i];      // merges into wide s_load
            tid[i] = token_type_ids[tok0 + i];
        }
#pragma unroll
        for (int i = 0; i < TPB; ++i)
            emit_token(wid[i], tid[i], pos0 + i, tok0 + i, h,
                       word_emb, pos_emb, type_emb, out);
    } else {
        // ---- tail path (not taken for S=512, TPB=8) ----
        for (int i = 0; i < TPB; ++i) {
            const int pos = pos0 + i;
            if (pos < seq_len) {
                emit_token(input_ids[tok0 + i], token_type_ids[tok0 + i],
                           pos, tok0 + i, h,
                           word_emb, pos_emb, type_emb, out);
            }
        }
    }
}

extern "C" void kernel_launch(void* const* d_in, const int* in_sizes, int n_in,
                              void* d_out, int out_size, void* d_ws, size_t ws_size,
                              hipStream_t stream) {
    (void)n_in; (void)d_ws; (void)ws_size; (void)out_size;

    const int*   input_ids      = (const int*)  d_in[0]; // [B*S]
    const int*   token_type_ids = (const int*)  d_in[1]; // [B*S]
    const float* word_emb       = (const float*)d_in[2]; // [VOCAB, H]
    const float* pos_emb        = (const float*)d_in[3]; // [MAX_POS, H]
    const float* type_emb       = (const float*)d_in[4]; // [TYPE_VOCAB, H]
    float*       out            = (float*)d_out;

    const int total_tokens = in_sizes[0];            // B*S = 16384
    const int seq_len      = in_sizes[3] / HIDDEN;   // MAX_POS == S == 512 here
    const int batch        = total_tokens / seq_len; // 32

    dim3 grid((seq_len + TPB - 1) / TPB, batch);     // (64, 32) = 2048 blocks
    dim3 block(VEC_PER_ROW);                         // 192 threads = 6 wave32

    bert_embedding_kernel<<<grid, block, 0, stream>>>(
        input_ids, token_type_ids, word_emb, pos_emb, type_emb, out, seq_len);
}